// GraphAttentionLayer_4810363372829
// MI455X (gfx1250) — compile-verified
//
#include <hip/hip_runtime.h>
#include <hip/hip_bf16.h>

// ---------------------------------------------------------------------------
// GAT layer for MI455X (gfx1250, wave32). fp32 end-to-end.
// GEMM stages use V_WMMA_F32_16X16X4_F32 (fp32 in/out, 16x16 tile per wave).
// ---------------------------------------------------------------------------

typedef __attribute__((ext_vector_type(2))) float v2f;
typedef __attribute__((ext_vector_type(8))) float v8f;

#define GAT_B 2
#define GAT_N 512
#define GAT_F 128
#define GAT_H 256   // 2F
#define NEG_SLOPE 0.04f
#define MASK_VAL  (-9.0e15f)

// ---------------------------------------------------------------------------
// Kernel 1: P = X * W1a^T, Q = X * W1b^T
//   X : (B*N, F) row-major        (1024 x 128)
//   w1: (H, H)   row-major        (256 x 256); W1a = w1[:, :F], W1b = w1[:, F:]
//   P,Q : (B*N, H)                (1024 x 256)
// One wave per 16x16 tile of P or Q. 2 * 64 * 16 = 2048 waves.
// ---------------------------------------------------------------------------
__global__ __launch_bounds__(256) void pq_gemm(const float* __restrict__ x,
                                               const float* __restrict__ w1,
                                               float* __restrict__ P,
                                               float* __restrict__ Q) {
    const int wave = (blockIdx.x * blockDim.x + threadIdx.x) >> 5;
    const int lane = threadIdx.x & 31;
    const int which = wave >> 10;          // 0 = P, 1 = Q
    const int rem   = wave & 1023;
    const int m0 = (rem >> 4) << 4;        // row tile over B*N = 1024
    const int n0 = (rem & 15) << 4;        // col tile over H = 256
    const int lhalf = lane >> 4;           // 0 or 1
    const int lm    = lane & 15;

    // A-frag source: x[m0+lm][k0 + 2*lhalf + v]
    const float* Arow = x + (size_t)(m0 + lm) * GAT_F;
    // B-frag source: B[k][n] = w1[n0+n][koff + k]  (koff = 0 for P, F for Q)
    const float* Brow = w1 + (size_t)(n0 + lm) * GAT_H + (which ? GAT_F : 0);

    v8f c = {0.f, 0.f, 0.f, 0.f, 0.f, 0.f, 0.f, 0.f};
    #pragma unroll 8
    for (int k0 = 0; k0 < GAT_F; k0 += 4) {
        const int kk = k0 + 2 * lhalf;
        v2f a = *(const v2f*)(Arow + kk);
        v2f b = *(const v2f*)(Brow + kk);
        c = __builtin_amdgcn_wmma_f32_16x16x4_f32(false, a, false, b,
                                                  (short)0, c, false, false);
    }

    // D layout: lanes 0-15 -> M = v, lanes 16-31 -> M = v + 8; N = lm
    float* out = (which ? Q : P);
    float* orow = out + (size_t)(m0 + 8 * lhalf) * GAT_H + n0 + lm;
    #pragma unroll
    for (int v = 0; v < 8; ++v) orow[(size_t)v * GAT_H] = c[v];
}

// ---------------------------------------------------------------------------
// Kernel 2: e[b,i,j] = leakyrelu( sum_h w2[h]*relu(P[b,i,h]+Q[b,j,h]+b1[h]) + b2 )
//           then masked by adj. One block per (b,i) row; 256 threads, 2 j each.
// ---------------------------------------------------------------------------
__global__ __launch_bounds__(256) void e_kernel(const float* __restrict__ P,
                                                const float* __restrict__ Q,
                                                const float* __restrict__ b1,
                                                const float* __restrict__ w2,
                                                const float* __restrict__ b2,
                                                const int*   __restrict__ adj,
                                                float* __restrict__ E) {
    const int bi = blockIdx.x;             // 0 .. B*N-1
    const int b  = bi >> 9;                // / N
    const int i  = bi & (GAT_N - 1);
    const int t  = threadIdx.x;

    __shared__ float s[GAT_H];             // P row + b1
    __shared__ float w2s[GAT_H];
    s[t]   = P[(size_t)bi * GAT_H + t] + b1[t];
    w2s[t] = w2[t];
    __syncthreads();

    const float b2v = b2[0];
    const int*   adjrow = adj + (size_t)b * GAT_N * GAT_N + (size_t)i * GAT_N;
    float*       erow   = E   + (size_t)b * GAT_N * GAT_N + (size_t)i * GAT_N;

    for (int j = t; j < GAT_N; j += 256) {
        const float* qrow = Q + (size_t)(b * GAT_N + j) * GAT_H;
        float acc = 0.f;
        #pragma unroll 4
        for (int h = 0; h < GAT_H; h += 4) {
            const float4 q4 = *(const float4*)(qrow + h);
            acc += w2s[h + 0] * fmaxf(s[h + 0] + q4.x, 0.f);
            acc += w2s[h + 1] * fmaxf(s[h + 1] + q4.y, 0.f);
            acc += w2s[h + 2] * fmaxf(s[h + 2] + q4.z, 0.f);
            acc += w2s[h + 3] * fmaxf(s[h + 3] + q4.w, 0.f);
        }
        float e = acc + b2v;
        e = (e > 0.f) ? e : NEG_SLOPE * e;
        e = (adjrow[j] > 0) ? e : MASK_VAL;
        erow[j] = e;
    }
}

// ---------------------------------------------------------------------------
// Kernel 3: row softmax over j (in place on E), and emit attention[0,0,:]
// One block per row (B*N rows), 256 threads, 2 elements each.
// ---------------------------------------------------------------------------
__global__ __launch_bounds__(256) void softmax_kernel(float* __restrict__ E,
                                                      float* __restrict__ att00) {
    const int row = blockIdx.x;
    const int t   = threadIdx.x;
    float* e = E + (size_t)row * GAT_N;

    __shared__ float red[256];

    const float x0 = e[t], x1 = e[t + 256];
    red[t] = fmaxf(x0, x1);
    __syncthreads();
    #pragma unroll
    for (int s = 128; s > 0; s >>= 1) {
        if (t < s) red[t] = fmaxf(red[t], red[t + s]);
        __syncthreads();
    }
    const float mx = red[0];
    __syncthreads();

    const float e0 = __expf(x0 - mx);
    const float e1 = __expf(x1 - mx);
    red[t] = e0 + e1;
    __syncthreads();
    #pragma unroll
    for (int s = 128; s > 0; s >>= 1) {
        if (t < s) red[t] += red[t + s];
        __syncthreads();
    }
    const float inv = 1.f / red[0];

    e[t]       = e0 * inv;
    e[t + 256] = e1 * inv;
    if (row == 0) {                       // b = 0, i = 0
        att00[t]       = e0 * inv;
        att00[t + 256] = e1 * inv;
    }
}

// ---------------------------------------------------------------------------
// Kernel 4: next_h[b] = att[b] (N x N) * x[b] (N x F), fp32 WMMA.
// One wave per 16x16 tile: 2 * 32 * 8 = 512 waves, K = 512.
// ---------------------------------------------------------------------------
__global__ __launch_bounds__(256) void av_gemm(const float* __restrict__ att,
                                               const float* __restrict__ x,
                                               float* __restrict__ out) {
    const int wave = (blockIdx.x * blockDim.x + threadIdx.x) >> 5;
    const int lane = threadIdx.x & 31;
    const int b    = wave >> 8;            // 0..1
    const int rem  = wave & 255;
    const int m0 = (rem >> 3) << 4;        // row tile over N
    const int n0 = (rem & 7) << 4;         // col tile over F
    const int lhalf = lane >> 4;
    const int lm    = lane & 15;

    const float* A = att + (size_t)b * GAT_N * GAT_N;
    const float* X = x   + (size_t)b * GAT_N * GAT_F;
    const float* Arow = A + (size_t)(m0 + lm) * GAT_N;   // contiguous in K
    const float* Bcol = X + n0 + lm;                     // strided in K

    v8f c = {0.f, 0.f, 0.f, 0.f, 0.f, 0.f, 0.f, 0.f};
    #pragma unroll 4
    for (int k0 = 0; k0 < GAT_N; k0 += 4) {
        const int kk = k0 + 2 * lhalf;
        v2f a = *(const v2f*)(Arow + kk);
        v2f bb;
        bb.x = Bcol[(size_t)kk * GAT_F];
        bb.y = Bcol[(size_t)(kk + 1) * GAT_F];
        c = __builtin_amdgcn_wmma_f32_16x16x4_f32(false, a, false, bb,
                                                  (short)0, c, false, false);
    }

    float* orow = out + (size_t)b * GAT_N * GAT_F
                      + (size_t)(m0 + 8 * lhalf) * GAT_F + n0 + lm;
    #pragma unroll
    for (int v = 0; v < 8; ++v) orow[(size_t)v * GAT_F] = c[v];
}

// ---------------------------------------------------------------------------
// Launch. Inputs in setup_inputs() order: x, w1, b1, w2, b2, adj.
// d_out = next_h (B*N*F floats) ++ attention[0,0,:] (N floats).
// Workspace: P (1MB) | Q (1MB) | E (2MB).
// ---------------------------------------------------------------------------
extern "C" void kernel_launch(void* const* d_in, const int* in_sizes, int n_in,
                              void* d_out, int out_size, void* d_ws, size_t ws_size,
                              hipStream_t stream) {
    const float* x   = (const float*)d_in[0];
    const float* w1  = (const float*)d_in[1];
    const float* b1  = (const float*)d_in[2];
    const float* w2  = (const float*)d_in[3];
    const float* b2  = (const float*)d_in[4];
    const int*   adj = (const int*)d_in[5];

    float* out = (float*)d_out;
    float* P = (float*)d_ws;                           // 1024*256
    float* Q = P + (size_t)GAT_B * GAT_N * GAT_H;      // 1024*256
    float* E = Q + (size_t)GAT_B * GAT_N * GAT_H;      // 2*512*512

    float* next_h = out;                                        // B*N*F
    float* att00  = out + (size_t)GAT_B * GAT_N * GAT_F;        // N

    // 1) P,Q projections: 2048 waves -> 256 blocks x 256 threads
    pq_gemm<<<256, 256, 0, stream>>>(x, w1, P, Q);
    // 2) attention logits with relu-MLP + leaky-relu + adjacency mask
    e_kernel<<<GAT_B * GAT_N, 256, 0, stream>>>(P, Q, b1, w2, b2, adj, E);
    // 3) row softmax (in place) + export attention[0,0,:]
    softmax_kernel<<<GAT_B * GAT_N, 256, 0, stream>>>(E, att00);
    // 4) next_h = att * x : 512 waves -> 64 blocks x 256 threads
    av_gemm<<<64, 256, 0, stream>>>(E, x, next_h);
}